// GIOULoss_80779744903189
// MI455X (gfx1250) — compile-verified
//
#include <hip/hip_runtime.h>
#include <hip/hip_bf16.h>

typedef __attribute__((ext_vector_type(2))) float v2f;
typedef __attribute__((ext_vector_type(8))) float v8f;

#define BATCH 32
#define KMAX  128
#define HH    512
#define WW    512
#define HW    (HH * WW)          // 262144
#define NELEM (BATCH * KMAX)     // 4096
#define NTHREADS 1024
#define ITEMS (NELEM / NTHREADS) // 4

__global__ __launch_bounds__(NTHREADS)
void GIOULoss_80779744903189_kernel(const float* __restrict__ owh,
                                    const float* __restrict__ org,
                                    const int* __restrict__ mask,
                                    const long long* __restrict__ ind,
                                    const float* __restrict__ twh,
                                    const float* __restrict__ trg,
                                    float* __restrict__ out) {
    const int tid = threadIdx.x;

    float lossAcc = 0.0f;
    float maskAcc = 0.0f;

#pragma unroll
    for (int i = 0; i < ITEMS; ++i) {
        const int e = tid + i * NTHREADS;            // element in [0, 4096)
        const int b = e >> 7;                         // e / K

        const long long iv = ind[e];                  // scatter index into H*W
        const int       mv = mask[e];

        // Gathers: channel stride is HW inside each batch image.
        const size_t base = (size_t)b * (size_t)(2 * HW) + (size_t)iv;
        const float pwx = owh[base];
        const float pwy = owh[base + HW];
        const float prx = org[base];
        const float pry = org[base + HW];

        const float twx = twh[2 * e];
        const float twy = twh[2 * e + 1];
        const float trx = trg[2 * e];
        const float trY = trg[2 * e + 1];

        const float xi = (float)(int)(iv & (long long)(WW - 1));
        const float yi = (float)(int)(iv >> 9);

        const float pcx = xi + prx, pcy = yi + pry;
        const float tcx = xi + trx, tcy = yi + trY;

        const float ax1 = pcx - pwx * 0.5f, ay1 = pcy - pwy * 0.5f;
        const float ax2 = pcx + pwx * 0.5f, ay2 = pcy + pwy * 0.5f;
        const float bx1 = tcx - twx * 0.5f, by1 = tcy - twy * 0.5f;
        const float bx2 = tcx + twx * 0.5f, by2 = tcy + twy * 0.5f;

        const float iw = fmaxf(fminf(ax2, bx2) - fmaxf(ax1, bx1), 0.0f);
        const float ih = fmaxf(fminf(ay2, by2) - fmaxf(ay1, by1), 0.0f);
        const float inter = iw * ih;
        const float uni = (ax2 - ax1) * (ay2 - ay1) +
                          (bx2 - bx1) * (by2 - by1) - inter;
        const float iou = inter / uni;
        const float cw = fmaxf(ax2, bx2) - fminf(ax1, bx1);
        const float ch = fmaxf(ay2, by2) - fminf(ay1, by1);
        const float ac = cw * ch;
        const float giou = iou - (ac - uni) / ac;

        const float mf = (mv > 0) ? 1.0f : 0.0f;
        lossAcc += (1.0f - giou) * mf;
        maskAcc += (float)mv;
    }

    // ---- wave32 reduction (deterministic, fixed order) ----
#pragma unroll
    for (int off = 16; off > 0; off >>= 1) {
        lossAcc += __shfl_xor(lossAcc, off, 32);
        maskAcc += __shfl_xor(maskAcc, off, 32);
    }

    __shared__ float sLoss[32];
    __shared__ float sMask[32];
    const int wave = tid >> 5;
    const int lane = tid & 31;
    if (lane == 0) {
        sLoss[wave] = lossAcc;
        sMask[wave] = maskAcc;
    }
    __syncthreads();

    // ---- last-mile 32 -> 1 reduction on the matrix pipe ----
    // A (16x4 f32) holds the 32 wave partials in a.x (a.y = 0); B = all-ones,
    // so D[m][n] = rowsum(A[m]) in exact fp32. Summing d0..d7 per lane and
    // folding lane^16 yields the total of all 64 A slots regardless of the
    // lane->(m,k) mapping. All 1024 threads execute this uniformly so EXEC
    // is all-1s for every wave (WMMA requirement).
    const float lp = sLoss[lane];
    const float mp = sMask[lane];

    v2f bOnes; bOnes.x = 1.0f; bOnes.y = 1.0f;
    v8f cZero = {0.f, 0.f, 0.f, 0.f, 0.f, 0.f, 0.f, 0.f};

    v2f aL; aL.x = lp; aL.y = 0.0f;
    v8f dL = __builtin_amdgcn_wmma_f32_16x16x4_f32(
        false, aL, false, bOnes, (short)0, cZero, false, false);
    float lt = dL[0] + dL[1] + dL[2] + dL[3] + dL[4] + dL[5] + dL[6] + dL[7];
    lt += __shfl_xor(lt, 16, 32);

    v2f aM; aM.x = mp; aM.y = 0.0f;
    v8f dM = __builtin_amdgcn_wmma_f32_16x16x4_f32(
        false, aM, false, bOnes, (short)0, cZero, false, false);
    float mt = dM[0] + dM[1] + dM[2] + dM[3] + dM[4] + dM[5] + dM[6] + dM[7];
    mt += __shfl_xor(mt, 16, 32);

    if (tid == 0) {
        out[0] = lt / (mt + 1e-4f);
    }
}

extern "C" void kernel_launch(void* const* d_in, const int* in_sizes, int n_in,
                              void* d_out, int out_size, void* d_ws, size_t ws_size,
                              hipStream_t stream) {
    (void)in_sizes; (void)n_in; (void)out_size; (void)d_ws; (void)ws_size;
    const float*     owh  = (const float*)d_in[0];      // output_wh (B,2,H,W) f32
    const float*     org  = (const float*)d_in[1];      // output_reg (B,2,H,W) f32
    const int*       mask = (const int*)d_in[2];        // (B,K) i32
    const long long* ind  = (const long long*)d_in[3];  // (B,K) i64
    const float*     twh  = (const float*)d_in[4];      // (B,K,2) f32
    const float*     trg  = (const float*)d_in[5];      // (B,K,2) f32
    float*           out  = (float*)d_out;              // scalar f32

    GIOULoss_80779744903189_kernel<<<dim3(1), dim3(NTHREADS), 0, stream>>>(
        owh, org, mask, ind, twh, trg, out);
}